// GAT_64029372449096
// MI455X (gfx1250) — compile-verified
//
#include <hip/hip_runtime.h>
#include <hip/hip_bf16.h>

#define HEADS   4
#define CH      128        // OUT_CH == IN_CH
#define HC      512        // HEADS * OUT_CH
#define NEG_SLOPE 0.2f

typedef float v2f __attribute__((ext_vector_type(2)));
typedef float v8f __attribute__((ext_vector_type(8)));

// ---------- order-preserving float<->uint encoding for atomic max ----------
__device__ __forceinline__ unsigned encF(float f) {
    unsigned u = __float_as_uint(f);
    return (u & 0x80000000u) ? ~u : (u | 0x80000000u);
}
__device__ __forceinline__ float decF(unsigned e) {
    unsigned b = (e & 0x80000000u) ? (e & 0x7FFFFFFFu) : ~e;
    return __uint_as_float(b);
}
__device__ __forceinline__ float lrelu(float a) {
    return a > 0.f ? a : NEG_SLOPE * a;
}

// ---------- 0) init: out = bias (broadcast), emax = encoded(-inf-ish), denom = 0 ----
__global__ void gat_init(float* __restrict__ out, const float* __restrict__ bias,
                         unsigned* __restrict__ emax, float* __restrict__ denom, int N) {
    long long i = (long long)blockIdx.x * blockDim.x + threadIdx.x;
    long long total = (long long)N * HC;
    if (i < total) out[i] = bias[i & (HC - 1)];
    if (i < (long long)N * HEADS) { emax[i] = 0u; denom[i] = 0.f; }
}

// ---------- 1) GEMM h = x @ W via V_WMMA_F32_16X16X4_F32 -----------------------
// one wave computes a 16(M) x 64(N) strip; 4 accumulators share each A fragment.
__global__ void gat_gemm_wmma(const float* __restrict__ X, const float* __restrict__ W,
                              float* __restrict__ H, int N) {
    int wid   = (blockIdx.x * blockDim.x + threadIdx.x) >> 5;   // global wave id
    int lane  = threadIdx.x & 31;
    int mt    = wid >> 3;                 // M tile (16 rows)
    int ns    = wid & 7;                  // N strip (64 cols of 512)
    int row0  = mt * 16;
    int col0  = ns * 64;
    if (row0 >= N) return;

    int half = lane >> 4;                 // 0: K0/K1, 1: K2/K3 (A & B layout)
    int l16  = lane & 15;

    int arow = row0 + l16;
    if (arow >= N) arow = N - 1;          // clamped duplicate (store guarded)
    const float* __restrict__ A = X + (size_t)arow * CH;

    v8f acc0 = {}, acc1 = {}, acc2 = {}, acc3 = {};

#pragma unroll 4
    for (int k0 = 0; k0 < CH; k0 += 4) {
        int ka = k0 + half * 2;
        v2f a; a.x = A[ka]; a.y = A[ka + 1];

        const float* __restrict__ B = W + (size_t)ka * HC + col0 + l16;
        v2f b0; b0.x = B[0];  b0.y = B[HC];
        v2f b1; b1.x = B[16]; b1.y = B[16 + HC];
        v2f b2; b2.x = B[32]; b2.y = B[32 + HC];
        v2f b3; b3.x = B[48]; b3.y = B[48 + HC];

        acc0 = __builtin_amdgcn_wmma_f32_16x16x4_f32(false, a, false, b0, (short)0, acc0, false, false);
        acc1 = __builtin_amdgcn_wmma_f32_16x16x4_f32(false, a, false, b1, (short)0, acc1, false, false);
        acc2 = __builtin_amdgcn_wmma_f32_16x16x4_f32(false, a, false, b2, (short)0, acc2, false, false);
        acc3 = __builtin_amdgcn_wmma_f32_16x16x4_f32(false, a, false, b3, (short)0, acc3, false, false);
    }

    // C/D layout: vgpr r -> row (r + 8*half), col = l16 within each 16-wide tile
#pragma unroll
    for (int r = 0; r < 8; ++r) {
        int m = row0 + half * 8 + r;
        if (m < N) {
            float* __restrict__ O = H + (size_t)m * HC + col0 + l16;
            O[0]  = acc0[r];
            O[16] = acc1[r];
            O[32] = acc2[r];
            O[48] = acc3[r];
        }
    }
}

// ---------- 2) per-node attention dots: a_src/a_dst [N, HEADS] -----------------
__global__ void gat_dots(const float* __restrict__ H, const float* __restrict__ att_src,
                         const float* __restrict__ att_dst, float* __restrict__ asrc,
                         float* __restrict__ adst, int N) {
    int n    = (blockIdx.x * blockDim.x + threadIdx.x) >> 5;   // node per wave
    int lane = threadIdx.x & 31;
    if (n >= N) return;
    const float4* __restrict__ hp = (const float4*)(H + (size_t)n * HC);
    const float4* __restrict__ s4 = (const float4*)att_src;
    const float4* __restrict__ d4 = (const float4*)att_dst;
    float ss[HEADS], sd[HEADS];
#pragma unroll
    for (int h = 0; h < HEADS; ++h) {
        float4 hv = hp[h * 32 + lane];
        float4 av = s4[h * 32 + lane];
        float4 bv = d4[h * 32 + lane];
        float s = hv.x * av.x + hv.y * av.y + hv.z * av.z + hv.w * av.w;
        float d = hv.x * bv.x + hv.y * bv.y + hv.z * bv.z + hv.w * bv.w;
#pragma unroll
        for (int m = 16; m > 0; m >>= 1) {
            s += __shfl_xor(s, m, 32);
            d += __shfl_xor(d, m, 32);
        }
        ss[h] = s; sd[h] = d;
    }
    if (lane == 0) {
#pragma unroll
        for (int h = 0; h < HEADS; ++h) {
            asrc[(size_t)n * HEADS + h] = ss[h];
            adst[(size_t)n * HEADS + h] = sd[h];
        }
    }
}

// ---------- 3) segment max over incoming edges (incl. self loops) --------------
__global__ void gat_edge_max(const int* __restrict__ src, const int* __restrict__ dst,
                             const float* __restrict__ asrc, const float* __restrict__ adst,
                             unsigned* __restrict__ emax, int E, int N) {
    int t = blockIdx.x * blockDim.x + threadIdx.x;
    if (t >= E + N) return;
    int s, d;
    if (t < E) { s = src[t]; d = dst[t]; } else { s = d = t - E; }
    float4 as = ((const float4*)asrc)[s];
    float4 ad = ((const float4*)adst)[d];
    float v[HEADS] = { as.x + ad.x, as.y + ad.y, as.z + ad.z, as.w + ad.w };
#pragma unroll
    for (int h = 0; h < HEADS; ++h)
        atomicMax(&emax[(size_t)d * HEADS + h], encF(lrelu(v[h])));
}

// ---------- 4) segment sum of exp(e - max) -------------------------------------
__global__ void gat_edge_expsum(const int* __restrict__ src, const int* __restrict__ dst,
                                const float* __restrict__ asrc, const float* __restrict__ adst,
                                const unsigned* __restrict__ emax, float* __restrict__ denom,
                                int E, int N) {
    int t = blockIdx.x * blockDim.x + threadIdx.x;
    if (t >= E + N) return;
    int s, d;
    if (t < E) { s = src[t]; d = dst[t]; } else { s = d = t - E; }
    float4 as = ((const float4*)asrc)[s];
    float4 ad = ((const float4*)adst)[d];
    float v[HEADS] = { as.x + ad.x, as.y + ad.y, as.z + ad.z, as.w + ad.w };
#pragma unroll
    for (int h = 0; h < HEADS; ++h) {
        float m = decF(emax[(size_t)d * HEADS + h]);
        atomicAdd(&denom[(size_t)d * HEADS + h], __expf(lrelu(v[h]) - m));
    }
}

// ---------- 5) weighted scatter: out[dst] += alpha * h[src] --------------------
// one wave per (edge, head); 32 lanes x float4 = 128 coalesced channels
__global__ void gat_scatter(const float* __restrict__ H, const int* __restrict__ src,
                            const int* __restrict__ dst, const float* __restrict__ asrc,
                            const float* __restrict__ adst, const unsigned* __restrict__ emax,
                            const float* __restrict__ denom, float* __restrict__ out,
                            int E, int N) {
    int wid  = (blockIdx.x * blockDim.x + threadIdx.x) >> 5;
    int lane = threadIdx.x & 31;
    int total = (E + N) * HEADS;
    if (wid >= total) return;
    int head = wid & (HEADS - 1);
    int e    = wid >> 2;
    int s, d;
    if (e < E) { s = src[e]; d = dst[e]; } else { s = d = e - E; }

    float a = lrelu(asrc[(size_t)s * HEADS + head] + adst[(size_t)d * HEADS + head]);
    float m = decF(emax[(size_t)d * HEADS + head]);
    float alpha = __expf(a - m) / denom[(size_t)d * HEADS + head];

    const float4* __restrict__ hp = (const float4*)(H + (size_t)s * HC + head * CH);
    float4 hv = hp[lane];
    float* __restrict__ op = out + (size_t)d * HC + head * CH + lane * 4;
    atomicAdd(op + 0, alpha * hv.x);
    atomicAdd(op + 1, alpha * hv.y);
    atomicAdd(op + 2, alpha * hv.z);
    atomicAdd(op + 3, alpha * hv.w);
}

extern "C" void kernel_launch(void* const* d_in, const int* in_sizes, int n_in,
                              void* d_out, int out_size, void* d_ws, size_t ws_size,
                              hipStream_t stream) {
    const float* x        = (const float*)d_in[0];
    const int*   eidx     = (const int*)d_in[1];     // [2, E] flat: src row then dst row
    const float* W        = (const float*)d_in[2];
    const float* att_src  = (const float*)d_in[3];
    const float* att_dst  = (const float*)d_in[4];
    const float* bias     = (const float*)d_in[5];
    float*       out      = (float*)d_out;

    const int N = in_sizes[0] / CH;
    const int E = in_sizes[1] / 2;
    const int* src = eidx;
    const int* dst = eidx + E;

    // workspace carve-up
    float*    H     = (float*)d_ws;                        // N*HC
    float*    asrc  = H + (size_t)N * HC;                  // N*HEADS
    float*    adst  = asrc + (size_t)N * HEADS;            // N*HEADS
    unsigned* emax  = (unsigned*)(adst + (size_t)N * HEADS); // N*HEADS
    float*    denom = (float*)(emax + (size_t)N * HEADS);  // N*HEADS

    const int TB = 256;

    // 0) out = bias broadcast; emax/denom reset
    {
        long long total = (long long)N * HC;
        int blocks = (int)((total + TB - 1) / TB);
        gat_init<<<blocks, TB, 0, stream>>>(out, bias, emax, denom, N);
    }
    // 1) h = x @ W  (WMMA fp32)
    {
        int mtiles = (N + 15) / 16;
        int waves  = mtiles * 8;              // 8 N-strips of 64 per M tile
        int blocks = (waves + 7) / 8;         // 8 waves per 256-thread block
        gat_gemm_wmma<<<blocks, TB, 0, stream>>>(x, W, H, N);
    }
    // 2) attention dots
    {
        long long threads = (long long)N * 32;
        int blocks = (int)((threads + TB - 1) / TB);
        gat_dots<<<blocks, TB, 0, stream>>>(H, att_src, att_dst, asrc, adst, N);
    }
    // 3) segment max
    {
        int blocks = (E + N + TB - 1) / TB;
        gat_edge_max<<<blocks, TB, 0, stream>>>(src, dst, asrc, adst, emax, E, N);
    }
    // 4) segment exp-sum
    {
        int blocks = (E + N + TB - 1) / TB;
        gat_edge_expsum<<<blocks, TB, 0, stream>>>(src, dst, asrc, adst, emax, denom, E, N);
    }
    // 5) weighted scatter into out
    {
        long long threads = (long long)(E + N) * HEADS * 32;
        int blocks = (int)((threads + TB - 1) / TB);
        gat_scatter<<<blocks, TB, 0, stream>>>(H, src, dst, asrc, adst, emax, denom, out, E, N);
    }
}